// CosineLoss_28484223107964
// MI455X (gfx1250) — compile-verified
//
#include <hip/hip_runtime.h>
#include <math.h>

typedef float v2f __attribute__((ext_vector_type(2)));
typedef float v8f __attribute__((ext_vector_type(8)));

#define EPS 1e-9f
#define NORM_FACTOR 0.1f
#define NUM_CLASSES 1000

// Stage 1: each wave32 computes the sum-of-squares of 16 rows via
// V_WMMA_F32_16X16X4_F32 with an all-ones B matrix (D[m][*] = row-sum of A),
// then lanes 0..7 / 16..23 gather pred[row, target[row]], normalize, and the
// block writes one (gather_sum, norm_loss_sum) partial pair.
__global__ __launch_bounds__(256) void cosine_rows_kernel(
    const float* __restrict__ pred, const int* __restrict__ target,
    float* __restrict__ partials, int nRows)
{
    const int C = NUM_CLASSES;
    const int lane = threadIdx.x & 31;
    const int wave = threadIdx.x >> 5;
    const long long baseRow = (long long)(blockIdx.x * 8 + wave) * 16;

    float g = 0.0f;   // normalized gathered value for this lane's row
    float nl = 0.0f;  // (1 - norm)^2 for this lane's row

    if (baseRow + 16 <= (long long)nRows) {
        const int rowInTile = lane & 15;   // which of the 16 rows this lane feeds
        const int half = lane >> 4;        // 0: K slots 0,1  1: K slots 2,3
        const float* rowPtr = pred + (size_t)(baseRow + rowInTile) * C + half * 4;

        v8f c = {0.f, 0.f, 0.f, 0.f, 0.f, 0.f, 0.f, 0.f};
        const v2f ones = {1.0f, 1.0f};

        // 1000 cols = 125 iterations * 8 cols. B == ones, so any permutation of
        // columns into K slots is valid: lane<16 covers cols k..k+3 of its row,
        // lane>=16 covers cols k+4..k+7.
        #pragma unroll 5
        for (int k = 0; k < C; k += 8) {
            __builtin_prefetch(rowPtr + k + 256, 0, 0);   // global_prefetch_b8
            float4 f = *(const float4*)(rowPtr + k);
            v2f a0 = { f.x * f.x, f.y * f.y };
            v2f a1 = { f.z * f.z, f.w * f.w };
            c = __builtin_amdgcn_wmma_f32_16x16x4_f32(
                    false, a0, false, ones, (short)0, c, false, false);
            c = __builtin_amdgcn_wmma_f32_16x16x4_f32(
                    false, a1, false, ones, (short)0, c, false, false);
        }

        // D layout (16x16 f32 C/D): lanes 0-15 hold rows 0..7 in c[0..7];
        // lanes 16-31 hold rows 8..15 in c[0..7]. Lanes 0..7 take rows 0..7,
        // lanes 16..23 take rows 8..15.
        if ((lane & 15) < 8) {
            const int idx = lane & 7;
            float ss = c[0];
            ss = (idx == 1) ? c[1] : ss;
            ss = (idx == 2) ? c[2] : ss;
            ss = (idx == 3) ? c[3] : ss;
            ss = (idx == 4) ? c[4] : ss;
            ss = (idx == 5) ? c[5] : ss;
            ss = (idx == 6) ? c[6] : ss;
            ss = (idx == 7) ? c[7] : ss;

            const long long row = baseRow + half * 8 + idx;
            const float norm = sqrtf(ss);
            const int t = target[row];
            const float p = pred[(size_t)row * C + (size_t)t];
            g = p / (norm + EPS);
            const float d = 1.0f - norm;
            nl = d * d;
        }
    }

    // wave32 reduction
    #pragma unroll
    for (int off = 16; off > 0; off >>= 1) {
        g  += __shfl_xor(g,  off, 32);
        nl += __shfl_xor(nl, off, 32);
    }

    __shared__ float sg[8];
    __shared__ float sn[8];
    if (lane == 0) { sg[wave] = g; sn[wave] = nl; }
    __syncthreads();
    if (threadIdx.x == 0) {
        float G = 0.f, NL = 0.f;
        #pragma unroll
        for (int i = 0; i < 8; ++i) { G += sg[i]; NL += sn[i]; }
        partials[2 * blockIdx.x]     = G;
        partials[2 * blockIdx.x + 1] = NL;
    }
}

// Stage 2: deterministic reduction of block partials -> scalar loss.
__global__ __launch_bounds__(256) void cosine_finalize_kernel(
    const float* __restrict__ partials, float* __restrict__ out,
    int nParts, float invN)
{
    float G = 0.f, NL = 0.f;
    for (int i = threadIdx.x; i < nParts; i += 256) {
        G  += partials[2 * i];
        NL += partials[2 * i + 1];
    }
    #pragma unroll
    for (int off = 16; off > 0; off >>= 1) {
        G  += __shfl_xor(G,  off, 32);
        NL += __shfl_xor(NL, off, 32);
    }
    __shared__ float sg[8];
    __shared__ float sn[8];
    const int lane = threadIdx.x & 31;
    const int wave = threadIdx.x >> 5;
    if (lane == 0) { sg[wave] = G; sn[wave] = NL; }
    __syncthreads();
    if (threadIdx.x == 0) {
        float aG = 0.f, aNL = 0.f;
        #pragma unroll
        for (int i = 0; i < 8; ++i) { aG += sg[i]; aNL += sn[i]; }
        out[0] = -aG * invN + NORM_FACTOR * (aNL * invN);
    }
}

extern "C" void kernel_launch(void* const* d_in, const int* in_sizes, int n_in,
                              void* d_out, int out_size, void* d_ws, size_t ws_size,
                              hipStream_t stream) {
    const float* pred   = (const float*)d_in[0];
    const int*   target = (const int*)d_in[1];
    float* out      = (float*)d_out;
    float* partials = (float*)d_ws;

    const int nRows = in_sizes[1];              // 65536 target entries = rows
    const int rowsPerBlock = 128;               // 8 waves * 16 rows
    const int grid = (nRows + rowsPerBlock - 1) / rowsPerBlock;   // 512

    cosine_rows_kernel<<<grid, 256, 0, stream>>>(pred, target, partials, nRows);
    cosine_finalize_kernel<<<1, 256, 0, stream>>>(partials, out, grid,
                                                  1.0f / (float)nRows);
}